// DistanceLayer_52020643889234
// MI455X (gfx1250) — compile-verified
//
#include <hip/hip_runtime.h>
#include <hip/hip_bf16.h>

// CDNA5 / gfx1250: fused similarity-mining kernel.
//  - fp32 GEMM tiles via V_WMMA_F32_16X16X4_F32 (sim matrix never materialized)
//  - global->LDS staging via GLOBAL_LOAD_ASYNC_TO_LDS_B128 (ASYNCcnt)
//  - branchless masked argmin/argmax with running (val,idx) trackers
// D=4 domains, B=8192 rows, F=128 features (fp32).
// Outputs (concat, fp32): pos_dist[D*B], neg_dist[D*B], pos_embed[D*B*F], neg_embed[D*B*F]

typedef float v2f __attribute__((ext_vector_type(2)));
typedef float v8f __attribute__((ext_vector_type(8)));

#define DOMS 4
#define BATCH 8192
#define FEAT 128
#define ROWS_PER_WG 128     // 8 waves x 16 rows
#define COLS_CHUNK 128      // columns staged in LDS per outer iteration
#define THREADS 256

__launch_bounds__(THREADS)
__global__ void mine_kernel(const float* __restrict__ emb,
                            const long long* __restrict__ labels,
                            float* __restrict__ pos_dist,
                            float* __restrict__ neg_dist,
                            float* __restrict__ pos_emb,
                            float* __restrict__ neg_emb)
{
    __shared__ int   s_lab[BATCH];                 // 32 KB: all labels of this domain
    __shared__ float s_tile[COLS_CHUNK * FEAT];    // 64 KB: staged column tile [c][k] == linear
    __shared__ int   s_pidx[ROWS_PER_WG];
    __shared__ int   s_nidx[ROWS_PER_WG];
    __shared__ int   s_hp[ROWS_PER_WG];

    const int bx     = blockIdx.x;
    const int d      = bx / (BATCH / ROWS_PER_WG);
    const int rowblk = (bx % (BATCH / ROWS_PER_WG)) * ROWS_PER_WG;
    const int tid    = threadIdx.x;
    const int wave   = tid >> 5;
    const int lane   = tid & 31;
    const int half   = lane >> 4;    // 0 or 1 (K-split for A/B, M-split for C/D)
    const int l16    = lane & 15;

    const float* demb = emb + (size_t)d * BATCH * FEAT;

    // Stage this domain's labels (int64 -> int32) into LDS once.
    for (int i = tid; i < BATCH; i += THREADS)
        s_lab[i] = (int)labels[(size_t)d * BATCH + i];
    __syncthreads();

    // This wave's 16-row strip.
    const int row0 = rowblk + wave * 16;

    // A fragments for V_WMMA_F32_16X16X4_F32:
    // lane holds row M = row0 + (lane%16); VGPR pair = K = 4*kk + 2*(lane/16) + {0,1}
    v2f afrag[FEAT / 4];
    {
        const float* arow = demb + (size_t)(row0 + l16) * FEAT;
#pragma unroll
        for (int kk = 0; kk < FEAT / 4; ++kk)
            afrag[kk] = *(const v2f*)(arow + 4 * kk + 2 * half);
    }

    // Labels of the 8 rows this lane's C-fragment covers (M = v + 8*half).
    int rlab[8];
#pragma unroll
    for (int v = 0; v < 8; ++v)
        rlab[v] = s_lab[row0 + 8 * half + v];

    // Running hardest-positive (min) / hardest-negative (max) trackers.
    float minv[8], maxv[8];
    int   mini[8], maxi[8];
#pragma unroll
    for (int v = 0; v < 8; ++v) {
        minv[v] =  __builtin_inff();
        maxv[v] = -__builtin_inff();
        mini[v] = 0x7fffffff;
        maxi[v] = 0x7fffffff;
    }

    // Async-staging addressing: the [c][k] tile layout over columns
    // chunk..chunk+127 is exactly the linear global layout, so the stage is a
    // straight 64 KB copy: 16 x (256 threads x 16B) fully-coalesced b128 ops.
    const uint32_t lds_tile_base = (uint32_t)(uintptr_t)&s_tile[0];
    const uint64_t gbase         = (uint64_t)(uintptr_t)demb;

    for (int chunk = 0; chunk < BATCH; chunk += COLS_CHUNK) {
        __syncthreads();   // previous tile fully consumed before overwrite
        {
            const uint32_t goff0 = (uint32_t)chunk * (FEAT * 4u);
#pragma unroll
            for (int i = 0; i < 16; ++i) {
                const uint32_t elem  = (uint32_t)(i * THREADS + tid) * 16u;
                const uint32_t laddr = lds_tile_base + elem;
                const uint32_t goff  = goff0 + elem;
                asm volatile("global_load_async_to_lds_b128 %0, %1, %2"
                             :: "v"(laddr), "v"(goff), "s"(gbase) : "memory");
            }
        }
        asm volatile("s_wait_asynccnt 0x0" ::: "memory");
        __syncthreads();

#pragma unroll 1
        for (int sub = 0; sub < COLS_CHUNK / 16; ++sub) {
            const int colg = chunk + sub * 16 + l16;   // this lane's column (N = lane%16)
            const int clab = s_lab[colg];

            v8f acc = {};
            const float* bcol = s_tile + (sub * 16 + l16) * FEAT + 2 * half;
#pragma unroll
            for (int kk = 0; kk < FEAT / 4; ++kk) {
                const v2f b = *(const v2f*)(bcol + 4 * kk);
                // D = A(16x4,f32) x B(4x16,f32) + C ; 8-arg form:
                // (neg_a, A, neg_b, B, c_mod, C, reuse_a, reuse_b)
                acc = __builtin_amdgcn_wmma_f32_16x16x4_f32(
                        false, afrag[kk], false, b, (short)0, acc, false, false);
            }

            // Branchless masked arg-min/arg-max update.
            // Lane holds sim[M = v+8*half][N = l16].
#pragma unroll
            for (int v = 0; v < 8; ++v) {
                const int   rowg = row0 + 8 * half + v;
                const float s    = acc[v];
                const bool  eq   = (rlab[v] == clab);

                const bool pt = (eq & (rowg != colg)) &
                                ((s < minv[v]) | ((s == minv[v]) & (colg < mini[v])));
                minv[v] = pt ? s    : minv[v];
                mini[v] = pt ? colg : mini[v];

                const bool nt = (!eq) &
                                ((s > maxv[v]) | ((s == maxv[v]) & (colg < maxi[v])));
                maxv[v] = nt ? s    : maxv[v];
                maxi[v] = nt ? colg : maxi[v];
            }
        }
    }

    // Reduce (val,idx) across the 16 lanes sharing each row set (xor within half-wave).
#pragma unroll
    for (int v = 0; v < 8; ++v) {
#pragma unroll
        for (int m = 1; m < 16; m <<= 1) {
            const float ov = __shfl_xor(minv[v], m, 32);
            const int   oi = __shfl_xor(mini[v], m, 32);
            const bool  tm = (ov < minv[v]) | ((ov == minv[v]) & (oi < mini[v]));
            minv[v] = tm ? ov : minv[v];
            mini[v] = tm ? oi : mini[v];

            const float ox  = __shfl_xor(maxv[v], m, 32);
            const int   oxi = __shfl_xor(maxi[v], m, 32);
            const bool  tx  = (ox > maxv[v]) | ((ox == maxv[v]) & (oxi < maxi[v]));
            maxv[v] = tx ? ox  : maxv[v];
            maxi[v] = tx ? oxi : maxi[v];
        }
    }

    // One writer per half-wave (lanes 0 and 16) emits the 8 rows it covers.
    if (l16 == 0) {
#pragma unroll
        for (int v = 0; v < 8; ++v) {
            const int rloc = wave * 16 + 8 * half + v;   // 0..127 within WG
            const int rowg = rowblk + rloc;
            const bool hp  = (minv[v] < __builtin_inff());  // has >=1 positive
            pos_dist[(size_t)d * BATCH + rowg] = hp ? minv[v] : 0.0f;
            neg_dist[(size_t)d * BATCH + rowg] = hp ? maxv[v] : 0.0f;
            s_pidx[rloc] = mini[v];
            s_nidx[rloc] = maxi[v];
            s_hp[rloc]   = hp ? 1 : 0;
        }
    }
    __syncthreads();

    // Gather phase: each wave copies the hardest-pos / hardest-neg embeddings
    // for its 16 rows (32 lanes x float4 = 128 floats per row), zeroed if !has_pos.
    for (int r = 0; r < 16; ++r) {
        const int rloc = wave * 16 + r;
        const int rowg = rowblk + rloc;
        float4 pv = {0.f, 0.f, 0.f, 0.f};
        float4 nv = {0.f, 0.f, 0.f, 0.f};
        if (s_hp[rloc]) {
            pv = *(const float4*)(demb + (size_t)s_pidx[rloc] * FEAT + lane * 4);
            nv = *(const float4*)(demb + (size_t)s_nidx[rloc] * FEAT + lane * 4);
        }
        *(float4*)(pos_emb + ((size_t)d * BATCH + rowg) * FEAT + lane * 4) = pv;
        *(float4*)(neg_emb + ((size_t)d * BATCH + rowg) * FEAT + lane * 4) = nv;
    }
}

extern "C" void kernel_launch(void* const* d_in, const int* in_sizes, int n_in,
                              void* d_out, int out_size, void* d_ws, size_t ws_size,
                              hipStream_t stream) {
    (void)in_sizes; (void)n_in; (void)d_ws; (void)ws_size; (void)out_size;

    const float*     semb = (const float*)d_in[0];
    const long long* slab = (const long long*)d_in[1];

    float* out      = (float*)d_out;
    float* pos_dist = out;                                   // [D*B]
    float* neg_dist = out + (size_t)DOMS * BATCH;            // [D*B]
    float* pos_emb  = out + (size_t)2 * DOMS * BATCH;        // [D*B*F]
    float* neg_emb  = pos_emb + (size_t)DOMS * BATCH * FEAT; // [D*B*F]

    dim3 grid(DOMS * (BATCH / ROWS_PER_WG));   // 4 * 64 = 256 workgroups
    dim3 block(THREADS);
    mine_kernel<<<grid, block, 0, stream>>>(semb, slab, pos_dist, neg_dist, pos_emb, neg_emb);
}